// Net_58445914964168
// MI455X (gfx1250) — compile-verified
//
#include <hip/hip_runtime.h>
#include <hip/hip_bf16.h>
#include <math.h>

#define NN 131072
#define NE 2097152

typedef float v2f __attribute__((ext_vector_type(2)));
typedef float v8f __attribute__((ext_vector_type(8)));

// ---------------------------------------------------------------------------
// degree accumulation on edge_index[0]
__global__ __launch_bounds__(256) void deg_kernel(const long long* __restrict__ ei,
                                                  float* __restrict__ deg) {
  int e = blockIdx.x * 256 + threadIdx.x;
  if (e < NE) {
    int s = (int)ei[e];
    atomicAdd(&deg[s], 1.0f);
  }
}

// deg -> dinv in place
__global__ __launch_bounds__(256) void dinv_kernel(float* __restrict__ deg) {
  int i = blockIdx.x * 256 + threadIdx.x;
  if (i < NN) {
    float d = deg[i];
    deg[i] = d > 0.f ? rsqrtf(d) : 0.f;
  }
}

// scalar (feature-dim 1) propagation: out[dst] += norm_e * xin[src]
__global__ __launch_bounds__(256) void prop_scalar_kernel(const long long* __restrict__ ei,
                                                          const float* __restrict__ dinv,
                                                          const float* __restrict__ xin,
                                                          float* __restrict__ out) {
  int e = blockIdx.x * 256 + threadIdx.x;
  if (e < NE) {
    int s = (int)ei[e];
    int d = (int)ei[NE + e];
    float w = -dinv[s] * dinv[d];
    atomicAdd(&out[d], w * xin[s]);
  }
}

// ChebConv layer 1 (in=1): pure per-node outer products + relu
__global__ __launch_bounds__(256) void layer1_kernel(const float* __restrict__ x,
                                                     const float* __restrict__ tx1,
                                                     const float* __restrict__ tx2p,
                                                     const float* __restrict__ W1,
                                                     const float* __restrict__ b1,
                                                     float* __restrict__ h1) {
  int t = blockIdx.x * 256 + threadIdx.x;  // over NN*64
  int node = t >> 6, c = t & 63;
  float xv  = x[node];
  float t1v = tx1[node];
  float t2v = 2.f * tx2p[node] - xv;
  float o = xv * W1[c] + t1v * W1[64 + c] + t2v * W1[128 + c] + b1[c];
  h1[t] = fmaxf(o, 0.f);
}

// 64-wide propagation: 16 lanes per edge, float4 per lane (coalesced gather +
// coalesced global_atomic_add_f32 scatter; h1/t1/t2p all L2-resident: 96MB of
// state in a 192MB L2, so the random traffic never reaches HBM)
__global__ __launch_bounds__(256) void prop_vec_kernel(const long long* __restrict__ ei,
                                                       const float* __restrict__ dinv,
                                                       const float* __restrict__ hin,
                                                       float* __restrict__ out) {
  unsigned g = blockIdx.x * 256u + threadIdx.x;
  unsigned e = g >> 4;
  unsigned sub = (g & 15u) * 4u;
  int s = (int)ei[e];
  int d = (int)ei[NE + e];
  float w = -dinv[s] * dinv[d];
  const float4 v = *(const float4*)(hin + (size_t)s * 64 + sub);
  float* o = out + (size_t)d * 64 + sub;
  atomicAdd(o + 0, w * v.x);
  atomicAdd(o + 1, w * v.y);
  atomicAdd(o + 2, w * v.z);
  atomicAdd(o + 3, w * v.w);
}

// ---------------------------------------------------------------------------
// ChebConv layer 2 (64->128) fully on the WMMA fp32 pipe.
//   out = h1@W2[0] + t1@W2[1] + (2*t2p - h1)@W2[2] + b2 ; relu ; row-max
// One wave = one 16-row tile, 8 accumulator tiles = all 128 output features,
// so relu+max fuse in registers and h2 [N,128] is never written to memory.
// W2 is staged in LDS as (k,k+1) float2 pairs so each B fragment is ONE
// aligned ds_load_b64 straight into an even VGPR pair feeding v_wmma.
__global__ __launch_bounds__(256) void layer2_wmma_kernel(const float* __restrict__ h1,
                                                          const float* __restrict__ t1,
                                                          const float* __restrict__ t2p,
                                                          const float* __restrict__ W2,
                                                          const float* __restrict__ b2,
                                                          float* __restrict__ colmax) {
  __shared__ float2 sW[32 * 128];  // 32 KB: one weight matrix, K pre-paired
  const int lane  = threadIdx.x & 31;
  const int wave  = threadIdx.x >> 5;
  const int l16   = lane & 15;
  const int halfk = (lane < 16) ? 0 : 2;     // A/B fragment K split across half-waves
  const int row0  = blockIdx.x * 128 + wave * 16;

  v8f acc[8];
#pragma unroll
  for (int nt = 0; nt < 8; ++nt)
#pragma unroll
    for (int r = 0; r < 8; ++r) acc[nt][r] = 0.f;

  for (int mat = 0; mat < 3; ++mat) {
    __syncthreads();  // protect sW against previous iteration's readers
    {
      const float* Wm = W2 + mat * 8192;
      for (int i = threadIdx.x; i < 32 * 128; i += 256) {
        int kp = i >> 7, col = i & 127;
        sW[i] = make_float2(Wm[(2 * kp) * 128 + col], Wm[(2 * kp + 1) * 128 + col]);
      }
    }
    __syncthreads();

    const float* A = (mat == 0) ? h1 : t1;  // mat==2 handled specially below

    for (int k0 = 0; k0 < 64; k0 += 4) {
      const int kb = k0 + halfk;            // even: 2-float A/B slices start here
      const size_t ra = (size_t)(row0 + l16) * 64 + kb;
      v2f a;
      if (mat == 2) {
        // Tx2 = 2*prop(Tx1) - h1, formed on the fly (no t2 buffer)
        const float2 p = *(const float2*)(t2p + ra);
        const float2 q = *(const float2*)(h1 + ra);
        a.x = 2.f * p.x - q.x;
        a.y = 2.f * p.y - q.y;
      } else {
        const float2 p = *(const float2*)(A + ra);
        a.x = p.x;
        a.y = p.y;
      }
      const int kprow = (kb >> 1) * 128;
#pragma unroll
      for (int nt = 0; nt < 8; ++nt) {
        const float2 bb = sW[kprow + nt * 16 + l16];  // single ds_load_b64
        v2f b;
        b.x = bb.x;
        b.y = bb.y;
        acc[nt] = __builtin_amdgcn_wmma_f32_16x16x4_f32(
            false, a, false, b, (short)0, acc[nt], false, false);
      }
    }
  }

  // epilogue: +bias, relu, row-wise max over the 128 features.
  // acc[nt][r] holds (row = row0 + r + (lane<16?0:8), col = nt*16 + l16)
#pragma unroll
  for (int r = 0; r < 8; ++r) {
    float m = 0.f;  // relu output is >= 0, so 0 is the identity for the max
#pragma unroll
    for (int nt = 0; nt < 8; ++nt) {
      float v = acc[nt][r] + b2[nt * 16 + l16];
      m = fmaxf(m, fmaxf(v, 0.f));
    }
    // max across the 16 lanes of each half-wave (xor bits 0..3 stay in-half)
    for (int off = 1; off < 16; off <<= 1) m = fmaxf(m, __shfl_xor(m, off, 32));
    if (l16 == 0) {
      int row = row0 + r + ((lane < 16) ? 0 : 8);
      colmax[row] = m;
    }
  }
}

// ---------------------------------------------------------------------------
// FC head: [1024,128] -> relu(fc1) -> fc2 -> [1024,1]  (tiny: 17 MFLOP)
__global__ __launch_bounds__(64) void fc_kernel(const float* __restrict__ colmax,
                                                const float* __restrict__ fc1w,
                                                const float* __restrict__ fc1b,
                                                const float* __restrict__ fc2w,
                                                const float* __restrict__ fc2b,
                                                float* __restrict__ out) {
  __shared__ float sh[128];
  __shared__ float red[64];
  const int row = blockIdx.x;  // 0..1023
  const int t = threadIdx.x;   // 0..63
  sh[t]      = colmax[row * 128 + t];
  sh[t + 64] = colmax[row * 128 + 64 + t];
  __syncthreads();
  float acc = fc1b[t];
  for (int k = 0; k < 128; ++k) acc += sh[k] * fc1w[k * 64 + t];
  red[t] = fmaxf(acc, 0.f) * fc2w[t];
  __syncthreads();
  if (t == 0) {
    float s = 0.f;
    for (int k = 0; k < 64; ++k) s += red[k];
    out[row] = s + fc2b[0];
  }
}

// ---------------------------------------------------------------------------
extern "C" void kernel_launch(void* const* d_in, const int* in_sizes, int n_in,
                              void* d_out, int out_size, void* d_ws, size_t ws_size,
                              hipStream_t stream) {
  (void)in_sizes; (void)n_in; (void)out_size; (void)ws_size;
  const float*     x    = (const float*)d_in[0];
  const long long* ei   = (const long long*)d_in[1];  // int64 in the reference
  const float*     W1   = (const float*)d_in[2];
  const float*     b1   = (const float*)d_in[3];
  const float*     W2   = (const float*)d_in[4];
  const float*     b2   = (const float*)d_in[5];
  const float*     fc1w = (const float*)d_in[6];
  const float*     fc1b = (const float*)d_in[7];
  const float*     fc2w = (const float*)d_in[8];
  const float*     fc2b = (const float*)d_in[9];
  float* out = (float*)d_out;

  // workspace layout (floats): dinv | tx1 | tx2p | h1(64N) | t1(64N) | t2p(64N) | colmax
  float* w    = (float*)d_ws;
  float* deg  = w;                       // becomes dinv in place
  float* tx1  = w + (size_t)NN;
  float* tx2p = w + (size_t)2 * NN;
  float* h1   = w + (size_t)3 * NN;
  float* t1   = w + (size_t)3 * NN + (size_t)64 * NN;
  float* t2p  = w + (size_t)3 * NN + (size_t)128 * NN;
  float* cmax = w + (size_t)3 * NN + (size_t)192 * NN;

  // zero all atomic-accumulated buffers (graph-capturable memsets)
  hipMemsetAsync(deg, 0, (size_t)3 * NN * sizeof(float), stream);
  hipMemsetAsync(t1, 0, (size_t)128 * NN * sizeof(float), stream);

  deg_kernel<<<NE / 256, 256, 0, stream>>>(ei, deg);
  dinv_kernel<<<NN / 256, 256, 0, stream>>>(deg);

  // layer 1 (feature dim 1)
  prop_scalar_kernel<<<NE / 256, 256, 0, stream>>>(ei, deg, x, tx1);
  prop_scalar_kernel<<<NE / 256, 256, 0, stream>>>(ei, deg, tx1, tx2p);
  layer1_kernel<<<(NN * 64) / 256, 256, 0, stream>>>(x, tx1, tx2p, W1, b1, h1);

  // layer 2 propagations (64-wide, L2-resident gather/atomic-scatter)
  prop_vec_kernel<<<(NE * 16) / 256, 256, 0, stream>>>(ei, deg, h1, t1);
  prop_vec_kernel<<<(NE * 16) / 256, 256, 0, stream>>>(ei, deg, t1, t2p);

  // layer 2 GEMMs + bias + relu + feature-max, fused, on fp32 WMMA
  layer2_wmma_kernel<<<NN / 128, 256, 0, stream>>>(h1, t1, t2p, W2, b2, cmax);

  // FC head -> [1024,1]
  fc_kernel<<<NN / 128, 64, 0, stream>>>(cmax, fc1w, fc1b, fc2w, fc2b, out);
}